// LocAttGrouped_20229295964635
// MI455X (gfx1250) — compile-verified
//
#include <hip/hip_runtime.h>
#include <hip/hip_bf16.h>

// ---------------------------------------------------------------------------
// Grouped local attention (3x3 window), MI455X / gfx1250.
//   Stage 1: q/k/v = 1x1 conv GEMMs via native fp32 WMMA (v_wmma_f32_16x16x4_f32)
//   Stage 2: per-pixel 9-neighbor attention via LDS-tiled VALU (memory bound)
// ---------------------------------------------------------------------------

typedef float v2f __attribute__((ext_vector_type(2)));
typedef float v8f __attribute__((ext_vector_type(8)));

#define CIN  256
#define COUT 256
#define HH   64
#define WW2  64
#define HW   (HH * WW2)      // 4096
#define NG   4
#define GD   64              // COUT / NG
#define CD   8               // d-chunk size for LDS staging in attention

// ===========================================================================
// Kernel 1: QKV projection GEMM.
//  grid = (HW/32 pixel tiles, 3 matrices, B batches), block = 256 (8 waves).
//  Wave w computes output tile [32 o-rows x 32 pixels] as 2x2 WMMA 16x16 tiles,
//  K-loop over Cin in steps of 4 (fp32 WMMA shape 16x16x4).
// ===========================================================================
__global__ __launch_bounds__(256) void qkv_gemm_kernel(
    const float* __restrict__ x,
    const float* __restrict__ wq, const float* __restrict__ bq,
    const float* __restrict__ wk, const float* __restrict__ bk,
    const float* __restrict__ wv, const float* __restrict__ bv,
    float* __restrict__ qout, float* __restrict__ kout, float* __restrict__ vout)
{
    const int m      = blockIdx.y;           // 0=q, 1=k, 2=v
    const int b      = blockIdx.z;
    const int p_base = blockIdx.x * 32;      // pixel tile within batch
    const int wave   = threadIdx.x >> 5;
    const int lane   = threadIdx.x & 31;
    const int o_base = wave * 32;

    const float* Wm = (m == 0) ? wq : (m == 1) ? wk : wv;
    const float* Bm = (m == 0) ? bq : (m == 1) ? bk : bv;
    float*       Om = (((m == 0) ? qout : (m == 1) ? kout : vout)) + (size_t)b * COUT * HW;
    const float* Xb = x + (size_t)b * CIN * HW;

    const int lo = lane & 15;   // 0..15
    const int hi = lane >> 4;   // 0 or 1

    v8f acc[2][2] = {};

    // A fragment (16x4 f32): lane L<16 -> row M=L, K=(k0,k0+1); L>=16 -> row M=L-16, K=(k0+2,k0+3)
    // B fragment (4x16 f32): v0 = row k0+2*hi at pixel lo (+16), v1 = row k0+2*hi+1
    #pragma unroll 4
    for (int k0 = 0; k0 < CIN; k0 += 4) {
        const int kc = k0 + 2 * hi;

        v2f a0 = *(const v2f*)(Wm + (size_t)(o_base + lo)      * CIN + kc);
        v2f a1 = *(const v2f*)(Wm + (size_t)(o_base + 16 + lo) * CIN + kc);

        const float* xp = Xb + (size_t)kc * HW + p_base + lo;
        v2f b0; b0.x = xp[0];  b0.y = xp[HW];
        v2f b1; b1.x = xp[16]; b1.y = xp[HW + 16];

        acc[0][0] = __builtin_amdgcn_wmma_f32_16x16x4_f32(false, a0, false, b0, (short)0, acc[0][0], false, false);
        acc[0][1] = __builtin_amdgcn_wmma_f32_16x16x4_f32(false, a0, false, b1, (short)0, acc[0][1], false, false);
        acc[1][0] = __builtin_amdgcn_wmma_f32_16x16x4_f32(false, a1, false, b0, (short)0, acc[1][0], false, false);
        acc[1][1] = __builtin_amdgcn_wmma_f32_16x16x4_f32(false, a1, false, b1, (short)0, acc[1][1], false, false);
    }

    // C/D layout: VGPR r -> M = r (lanes 0-15) / r+8 (lanes 16-31); N = lane&15.
    #pragma unroll
    for (int ot = 0; ot < 2; ++ot) {
        #pragma unroll
        for (int pt = 0; pt < 2; ++pt) {
            #pragma unroll
            for (int r = 0; r < 8; ++r) {
                const int o = o_base + ot * 16 + r + hi * 8;
                const int p = p_base + pt * 16 + lo;
                Om[(size_t)o * HW + p] = acc[ot][pt][r] + Bm[o];
            }
        }
    }
}

// ===========================================================================
// Kernel 2: grouped 3x3 local attention over q/k/v.
//  grid = (W/16, H/16, B*G), block = (16,16). LDS stages k (then v) with a
//  1-pixel halo in d-chunks of CD=8: 18*18*8 floats = 10.4 KB.
// ===========================================================================
__global__ __launch_bounds__(256) void local_attn_kernel(
    const float* __restrict__ q, const float* __restrict__ k,
    const float* __restrict__ v, float* __restrict__ out)
{
    __shared__ float smem[18 * 18 * CD];

    const int tx = threadIdx.x, ty = threadIdx.y;
    const int w0 = blockIdx.x * 16, h0 = blockIdx.y * 16;
    const int b  = blockIdx.z >> 2;
    const int g  = blockIdx.z & 3;
    const int h  = h0 + ty, w = w0 + tx;
    const int tid = ty * 16 + tx;
    const int cbase = b * COUT + g * GD;   // channel-plane base index

    float att[9];
    #pragma unroll
    for (int i = 0; i < 9; ++i) att[i] = 0.f;

    // ---- pass 1: attn[p] = sum_d q[d] * k_nb[p][d] ----
    for (int dc = 0; dc < GD / CD; ++dc) {
        for (int i = tid; i < 18 * 18 * CD; i += 256) {
            const int dd  = i / 324;
            const int rem = i - dd * 324;
            const int y   = rem / 18;
            const int xx  = rem - y * 18;
            const int gh  = h0 - 1 + y, gw = w0 - 1 + xx;
            float val = 0.f;
            if (gh >= 0 && gh < HH && gw >= 0 && gw < WW2)
                val = k[(size_t)(cbase + dc * CD + dd) * HW + gh * WW2 + gw];
            smem[i] = val;
        }
        __syncthreads();
        #pragma unroll
        for (int dd = 0; dd < CD; ++dd) {
            const float qv = q[(size_t)(cbase + dc * CD + dd) * HW + h * WW2 + w];
            const float* sp = smem + dd * 324;
            #pragma unroll
            for (int dy = 0; dy < 3; ++dy)
                #pragma unroll
                for (int dx = 0; dx < 3; ++dx)
                    att[dy * 3 + dx] += qv * sp[(ty + dy) * 18 + (tx + dx)];
        }
        __syncthreads();
    }

    // ---- softmax over the 9 window positions ----
    float mx = att[0];
    #pragma unroll
    for (int i = 1; i < 9; ++i) mx = fmaxf(mx, att[i]);
    float s = 0.f;
    #pragma unroll
    for (int i = 0; i < 9; ++i) { att[i] = __expf(att[i] - mx); s += att[i]; }
    const float inv = 1.0f / s;
    #pragma unroll
    for (int i = 0; i < 9; ++i) att[i] *= inv;

    // ---- pass 2: out[d] = sum_p attn[p] * v_nb[p][d] ----
    for (int dc = 0; dc < GD / CD; ++dc) {
        for (int i = tid; i < 18 * 18 * CD; i += 256) {
            const int dd  = i / 324;
            const int rem = i - dd * 324;
            const int y   = rem / 18;
            const int xx  = rem - y * 18;
            const int gh  = h0 - 1 + y, gw = w0 - 1 + xx;
            float val = 0.f;
            if (gh >= 0 && gh < HH && gw >= 0 && gw < WW2)
                val = v[(size_t)(cbase + dc * CD + dd) * HW + gh * WW2 + gw];
            smem[i] = val;
        }
        __syncthreads();
        #pragma unroll
        for (int dd = 0; dd < CD; ++dd) {
            const float* sp = smem + dd * 324;
            float o = 0.f;
            #pragma unroll
            for (int dy = 0; dy < 3; ++dy)
                #pragma unroll
                for (int dx = 0; dx < 3; ++dx)
                    o += att[dy * 3 + dx] * sp[(ty + dy) * 18 + (tx + dx)];
            out[(size_t)(cbase + dc * CD + dd) * HW + h * WW2 + w] = o;
        }
        __syncthreads();
    }
}

// ===========================================================================
extern "C" void kernel_launch(void* const* d_in, const int* in_sizes, int n_in,
                              void* d_out, int out_size, void* d_ws, size_t ws_size,
                              hipStream_t stream) {
    const float* x  = (const float*)d_in[0];
    const float* wq = (const float*)d_in[1];
    const float* bq = (const float*)d_in[2];
    const float* wk = (const float*)d_in[3];
    const float* bk = (const float*)d_in[4];
    const float* wv = (const float*)d_in[5];
    const float* bv = (const float*)d_in[6];
    float* out = (float*)d_out;

    const size_t planes = (size_t)4 * COUT * HW;   // 4,194,304 floats per matrix
    float* qb = (float*)d_ws;
    float* kb = qb + planes;
    float* vb = kb + planes;

    // Stage 1: q/k/v GEMMs. 128 pixel-tiles x 3 matrices x 4 batches.
    qkv_gemm_kernel<<<dim3(HW / 32, 3, 4), 256, 0, stream>>>(
        x, wq, bq, wk, bk, wv, bv, qb, kb, vb);

    // Stage 2: local attention. 4x4 spatial tiles x (B*G)=16.
    local_attn_kernel<<<dim3(WW2 / 16, HH / 16, 16), dim3(16, 16), 0, stream>>>(
        qb, kb, vb, out);
}